// MeshEncoder_5385888989266
// MI455X (gfx1250) — compile-verified
//
#include <hip/hip_runtime.h>
#include <hip/hip_bf16.h>
#include <math.h>

typedef float v2f __attribute__((ext_vector_type(2)));
typedef float v8f __attribute__((ext_vector_type(8)));

#define HID 256
#define HEADS 4

__device__ __forceinline__ float lrelu(float x) { return x > 0.0f ? x : 0.2f * x; }

// order-preserving float <-> uint encoding for atomicMax on floats
__device__ __forceinline__ unsigned fenc(float f) {
    unsigned u = __float_as_uint(f);
    return (u & 0x80000000u) ? ~u : (u | 0x80000000u);
}
__device__ __forceinline__ float fdec(unsigned u) {
    return __uint_as_float((u & 0x80000000u) ? (u ^ 0x80000000u) : ~u);
}

// ---------------- generic fill ----------------
__global__ void k_fill(float* __restrict__ p, float v, long long n) {
    long long i = (long long)blockIdx.x * blockDim.x + threadIdx.x;
    long long stride = (long long)gridDim.x * blockDim.x;
    for (; i < n; i += stride) p[i] = v;
}

// ---------------- degree / norm ----------------
__global__ void k_deg_edges(const int* __restrict__ dst, float* __restrict__ deg, int E) {
    int e = blockIdx.x * blockDim.x + threadIdx.x;
    if (e < E) atomicAdd(&deg[dst[e]], 1.0f);
}
__global__ void k_rsqrt(float* __restrict__ d, int N) {
    int i = blockIdx.x * blockDim.x + threadIdx.x;
    if (i < N) { float v = d[i]; d[i] = (v > 0.0f) ? rsqrtf(v) : 0.0f; }
}

// ---------------- GCN layer 1 input GEMM (K=3, scalar) ----------------
__global__ void k_lin3(const float* __restrict__ x, const float* __restrict__ W1,
                       float* __restrict__ P, int N) {
    int idx = blockIdx.x * blockDim.x + threadIdx.x;
    if (idx >= N * 64) return;
    int i = idx >> 6, c = idx & 63;
    P[idx] = x[i * 3 + 0] * W1[c] + x[i * 3 + 1] * W1[64 + c] + x[i * 3 + 2] * W1[128 + c];
}

// ---------------- GCN edge aggregation (thread per edge-feature) ----------------
__global__ void k_gcn_agg(const int* __restrict__ src, const int* __restrict__ dst,
                          const float* __restrict__ dis, const float* __restrict__ P,
                          float* __restrict__ AGG, int E, int logF) {
    long long t = (long long)blockIdx.x * blockDim.x + threadIdx.x;
    if (t >= ((long long)E << logF)) return;
    int e = (int)(t >> logF);
    int f = (int)(t & ((1 << logF) - 1));
    int s = src[e], d = dst[e];
    float nrm = dis[s] * dis[d];
    atomicAdd(&AGG[((long long)d << logF) + f], P[((long long)s << logF) + f] * nrm);
}

// self-loop + bias + relu (in place: H == AGG)
__global__ void k_gcn_finish(const float* __restrict__ P, const float* __restrict__ dis,
                             const float* __restrict__ b, float* __restrict__ H,
                             int N, int logF) {
    long long idx = (long long)blockIdx.x * blockDim.x + threadIdx.x;
    if (idx >= ((long long)N << logF)) return;
    int i = (int)(idx >> logF);
    int f = (int)(idx & ((1 << logF) - 1));
    float dd = dis[i];
    float v = H[idx] + P[idx] * dd * dd + b[f];
    H[idx] = fmaxf(v, 0.0f);
}

// ---------------- f32 WMMA tiled GEMM: C[M x Ncols] = A[M x K] * B[K x Ncols] ----------------
// wave computes one 16x16 tile via V_WMMA_F32_16X16X4_F32 (exact fp32)
__global__ void k_gemm_wmma_f32(const float* __restrict__ A, int lda,
                                const float* __restrict__ B, int ldb, int bcol,
                                float* __restrict__ C, int ldc,
                                const float* __restrict__ bias, int doRelu,
                                int M, int Ncols, int K) {
    int wave = (blockIdx.x * blockDim.x + threadIdx.x) >> 5;
    int lane = threadIdx.x & 31;
    int tilesN = Ncols >> 4;
    int tilesM = M >> 4;
    if (wave >= tilesM * tilesN) return;
    int tm = (wave / tilesN) << 4;
    int tn = (wave % tilesN) << 4;
    int r = lane & 15;      // row-in-tile (A) / col-in-tile (B,C)
    int half = lane >> 4;   // K split for A/B, M split for C/D

    v8f acc = {};
    const float* Arow = A + (long long)(tm + r) * lda + (half << 1);
    const float* Bp = B + bcol + tn + r;
#pragma unroll 4
    for (int k = 0; k < K; k += 4) {
        v2f a, b;
        a.x = Arow[k];
        a.y = Arow[k + 1];
        int kb = k + (half << 1);
        b.x = Bp[(long long)kb * ldb];
        b.y = Bp[(long long)(kb + 1) * ldb];
        acc = __builtin_amdgcn_wmma_f32_16x16x4_f32(false, a, false, b, (short)0, acc,
                                                    false, false);
    }
    float bv = bias ? bias[tn + r] : 0.0f;
#pragma unroll
    for (int i = 0; i < 8; ++i) {
        int row = tm + i + (half << 3);
        float v = acc[i] + bv;
        if (doRelu) v = fmaxf(v, 0.0f);
        C[(long long)row * ldc + tn + r] = v;
    }
}

// ---------------- attention scores: s = hk . a  (wave per node) ----------------
__global__ void k_scores(const float* __restrict__ HK, const float* __restrict__ asrc,
                         const float* __restrict__ adst, float* __restrict__ ssrc,
                         float* __restrict__ sdst, int N) {
    int wave = (blockIdx.x * blockDim.x + threadIdx.x) >> 5;
    int lane = threadIdx.x & 31;
    if (wave >= N) return;
    const float* row = HK + (long long)wave * HID;
    float s1 = 0.0f, s2 = 0.0f;
#pragma unroll
    for (int j = lane; j < HID; j += 32) {
        float h = row[j];
        s1 += h * asrc[j];
        s2 += h * adst[j];
    }
    for (int m = 16; m > 0; m >>= 1) {
        s1 += __shfl_xor(s1, m, 32);
        s2 += __shfl_xor(s2, m, 32);
    }
    if (lane == 0) { ssrc[wave] = s1; sdst[wave] = s2; }
}

__global__ void k_att_init(const float* __restrict__ ssrc, const float* __restrict__ sdst,
                           unsigned* __restrict__ emaxU, int N) {
    int i = blockIdx.x * blockDim.x + threadIdx.x;
    if (i < N) emaxU[i] = fenc(lrelu(ssrc[i] + sdst[i]));   // self-loop edge value
}
__global__ void k_att_emax(const int* __restrict__ src, const int* __restrict__ dst,
                           const float* __restrict__ ssrc, const float* __restrict__ sdst,
                           unsigned* __restrict__ emaxU, int E) {
    int e = blockIdx.x * blockDim.x + threadIdx.x;
    if (e >= E) return;
    float ev = lrelu(ssrc[src[e]] + sdst[dst[e]]);
    atomicMax(&emaxU[dst[e]], fenc(ev));
}
__global__ void k_att_denom_init(const float* __restrict__ ssrc, const float* __restrict__ sdst,
                                 const unsigned* __restrict__ emaxU, float* __restrict__ emaxF,
                                 float* __restrict__ denom, int N) {
    int i = blockIdx.x * blockDim.x + threadIdx.x;
    if (i >= N) return;
    float m = fdec(emaxU[i]);
    emaxF[i] = m;
    denom[i] = expf(lrelu(ssrc[i] + sdst[i]) - m);          // self-loop term
}
__global__ void k_att_ee(const int* __restrict__ src, const int* __restrict__ dst,
                         const float* __restrict__ ssrc, const float* __restrict__ sdst,
                         const float* __restrict__ emaxF, float* __restrict__ ee,
                         float* __restrict__ denom, int E) {
    int e = blockIdx.x * blockDim.x + threadIdx.x;
    if (e >= E) return;
    int d = dst[e];
    float v = expf(lrelu(ssrc[src[e]] + sdst[d]) - emaxF[d]);
    ee[e] = v;
    atomicAdd(&denom[d], v);
}
// weighted scatter (wave per edge, coalesced feature chunks)
__global__ void k_att_scatter(const int* __restrict__ src, const int* __restrict__ dst,
                              const float* __restrict__ ee, const float* __restrict__ denom,
                              const float* __restrict__ HK, float* __restrict__ OUT, int E) {
    int wave = (blockIdx.x * blockDim.x + threadIdx.x) >> 5;
    int lane = threadIdx.x & 31;
    if (wave >= E) return;
    int s = src[wave], d = dst[wave];
    float coef = ee[wave] / (denom[d] + 1e-16f);
    const float* hs = HK + (long long)s * HID;
    float* od = OUT + (long long)d * HID;
#pragma unroll
    for (int j = lane; j < HID; j += 32) atomicAdd(&od[j], coef * hs[j]);
}
__global__ void k_att_self(const float* __restrict__ ssrc, const float* __restrict__ sdst,
                           const float* __restrict__ emaxF, const float* __restrict__ denom,
                           const float* __restrict__ HK, float* __restrict__ OUT, int N) {
    int wave = (blockIdx.x * blockDim.x + threadIdx.x) >> 5;
    int lane = threadIdx.x & 31;
    if (wave >= N) return;
    float es = expf(lrelu(ssrc[wave] + sdst[wave]) - emaxF[wave]);
    float coef = es / (denom[wave] + 1e-16f);
    const float* hs = HK + (long long)wave * HID;
    float* od = OUT + (long long)wave * HID;
#pragma unroll
    for (int j = lane; j < HID; j += 32) od[j] += coef * hs[j];
}

// x3 = OUT / HEADS + b3   (written straight into d_out's x3 region)
__global__ void k_x3(const float* __restrict__ OUT, const float* __restrict__ b3,
                     float* __restrict__ x3, int N) {
    long long idx = (long long)blockIdx.x * blockDim.x + threadIdx.x;
    if (idx >= (long long)N * HID) return;
    int f = (int)(idx & (HID - 1));
    x3[idx] = OUT[idx] * (1.0f / HEADS) + b3[f];
}

// gate[i] = T[i,:] . Gw2 + Gb2  (wave per node)
__global__ void k_rowdot(const float* __restrict__ T, const float* __restrict__ Gw2,
                         const float* __restrict__ Gb2, float* __restrict__ gate, int N) {
    int wave = (blockIdx.x * blockDim.x + threadIdx.x) >> 5;
    int lane = threadIdx.x & 31;
    if (wave >= N) return;
    const float* row = T + (long long)wave * HID;
    float s = 0.0f;
#pragma unroll
    for (int j = lane; j < HID; j += 32) s += row[j] * Gw2[j];
    for (int m = 16; m > 0; m >>= 1) s += __shfl_xor(s, m, 32);
    if (lane == 0) gate[wave] = s + Gb2[0];
}

__global__ void k_scal_init(unsigned* __restrict__ scal) {
    if (threadIdx.x == 0) {
        scal[0] = fenc(-3.4e38f);            // running max (encoded)
        ((float*)scal)[1] = 0.0f;            // running sum
    }
}
__global__ void k_gmax(const float* __restrict__ gate, unsigned* __restrict__ scal, int N) {
    int i = blockIdx.x * blockDim.x + threadIdx.x;
    float v = (i < N) ? gate[i] : -3.4e38f;
    for (int m = 16; m > 0; m >>= 1) v = fmaxf(v, __shfl_xor(v, m, 32));
    if ((threadIdx.x & 31) == 0) atomicMax(&scal[0], fenc(v));
}
__global__ void k_wts(const float* __restrict__ gate, unsigned* __restrict__ scal,
                      float* __restrict__ wts, int N) {
    int i = blockIdx.x * blockDim.x + threadIdx.x;
    float gmax = fdec(scal[0]);
    float w = (i < N) ? expf(gate[i] - gmax) : 0.0f;
    if (i < N) wts[i] = w;
    float s = w;
    for (int m = 16; m > 0; m >>= 1) s += __shfl_xor(s, m, 32);
    if ((threadIdx.x & 31) == 0) atomicAdd(&((float*)scal)[1], s);
}
// pooled g accumulation: block handles 128 nodes, thread f accumulates feature f
__global__ void k_gacc(const float* __restrict__ wts, const unsigned* __restrict__ scal,
                       const float* __restrict__ x3, float* __restrict__ gacc, int N) {
    int f = threadIdx.x;
    int n0 = blockIdx.x * 128;
    int n1 = min(n0 + 128, N);
    float gsum = ((const float*)scal)[1];
    float acc = 0.0f;
    for (int n = n0; n < n1; ++n) acc += (wts[n] / gsum) * x3[(long long)n * HID + f];
    atomicAdd(&gacc[f], acc);
}
__global__ void k_gfinal(const float* __restrict__ gacc, const float* __restrict__ Mw,
                         const float* __restrict__ Mb, float* __restrict__ g) {
    int c = threadIdx.x;
    float s = 0.0f;
#pragma unroll 4
    for (int j = 0; j < HID; ++j) s += gacc[j] * Mw[j * HID + c];
    g[c] = fmaxf(s + Mb[c], 0.0f);
}

extern "C" void kernel_launch(void* const* d_in, const int* in_sizes, int n_in,
                              void* d_out, int out_size, void* d_ws, size_t ws_size,
                              hipStream_t stream) {
    const float* x    = (const float*)d_in[0];
    const int*   ei   = (const int*)d_in[1];
    const float* W1   = (const float*)d_in[2];
    const float* b1   = (const float*)d_in[3];
    const float* W2   = (const float*)d_in[4];
    const float* b2   = (const float*)d_in[5];
    const float* W3   = (const float*)d_in[6];
    const float* b3   = (const float*)d_in[7];
    const float* aS   = (const float*)d_in[8];
    const float* aD   = (const float*)d_in[9];
    const float* Gw1  = (const float*)d_in[10];
    const float* Gb1  = (const float*)d_in[11];
    const float* Gw2  = (const float*)d_in[12];
    const float* Gb2  = (const float*)d_in[13];
    const float* Mw   = (const float*)d_in[14];
    const float* Mb   = (const float*)d_in[15];

    const int N = in_sizes[0] / 3;
    const int E = in_sizes[1] / 2;
    const int* src = ei;
    const int* dst = ei + E;

    // ---- workspace layout (floats) ----
    float* ws = (float*)d_ws;
    long long off = 0;
    float* dis  = ws + off; off += N;                    // degree -> 1/sqrt(deg)
    float* P    = ws + off; off += (long long)N * 128;   // pre-agg features (p1 then p2)
    float* H    = ws + off; off += (long long)N * 128;   // agg/h1 then h2
    float* HK   = ws + off; off += (long long)N * HID;   // per-head features
    float* OUT  = ws + off; off += (long long)N * HID;   // head accumulation; reused as T
    float* ssrc = ws + off; off += N;
    float* sdst = ws + off; off += N;
    unsigned* emaxU = (unsigned*)(ws + off); off += N;
    float* emaxF = ws + off; off += N;
    float* denom = ws + off; off += N;
    float* ee    = ws + off; off += E;
    float* gate  = ws + off; off += N;
    float* wts   = ws + off; off += N;
    float* gacc  = ws + off; off += HID;
    unsigned* scal = (unsigned*)(ws + off); off += 8;

    float* x3 = (float*)d_out + HID;   // output tuple: g[256] then x3[N*256]
    float* gO = (float*)d_out;

    const int B = 256;
    auto cdiv = [](long long a, long long b) { return (int)((a + b - 1) / b); };

    // ---- degrees / symmetric norm ----
    k_fill<<<cdiv(N, B), B, 0, stream>>>(dis, 1.0f, N);                  // self-loop
    k_deg_edges<<<cdiv(E, B), B, 0, stream>>>(dst, dis, E);
    k_rsqrt<<<cdiv(N, B), B, 0, stream>>>(dis, N);

    // ---- GCN layer 1: x@W1 -> scatter -> relu ----
    k_lin3<<<cdiv((long long)N * 64, B), B, 0, stream>>>(x, W1, P, N);
    k_fill<<<cdiv((long long)N * 64, B), B, 0, stream>>>(H, 0.0f, (long long)N * 64);
    k_gcn_agg<<<cdiv((long long)E * 64, B), B, 0, stream>>>(src, dst, dis, P, H, E, 6);
    k_gcn_finish<<<cdiv((long long)N * 64, B), B, 0, stream>>>(P, dis, b1, H, N, 6);

    // ---- GCN layer 2: h1@W2 (WMMA) -> scatter -> relu ----
    {
        int waves = (N / 16) * (128 / 16);
        k_gemm_wmma_f32<<<cdiv(waves * 32, B), B, 0, stream>>>(H, 64, W2, 128, 0, P, 128,
                                                               nullptr, 0, N, 128, 64);
    }
    k_fill<<<cdiv((long long)N * 128, B), B, 0, stream>>>(H, 0.0f, (long long)N * 128);
    k_gcn_agg<<<cdiv((long long)E * 128, B), B, 0, stream>>>(src, dst, dis, P, H, E, 7);
    k_gcn_finish<<<cdiv((long long)N * 128, B), B, 0, stream>>>(P, dis, b2, H, N, 7);

    // ---- GAT heads ----
    k_fill<<<cdiv((long long)N * HID, B), B, 0, stream>>>(OUT, 0.0f, (long long)N * HID);
    for (int k = 0; k < HEADS; ++k) {
        int waves = (N / 16) * (HID / 16);
        k_gemm_wmma_f32<<<cdiv(waves * 32, B), B, 0, stream>>>(
            H, 128, W3, HEADS * HID, k * HID, HK, HID, nullptr, 0, N, HID, 128);
        k_scores<<<cdiv((long long)N * 32, B), B, 0, stream>>>(HK, aS + k * HID, aD + k * HID,
                                                               ssrc, sdst, N);
        k_att_init<<<cdiv(N, B), B, 0, stream>>>(ssrc, sdst, emaxU, N);
        k_att_emax<<<cdiv(E, B), B, 0, stream>>>(src, dst, ssrc, sdst, emaxU, E);
        k_att_denom_init<<<cdiv(N, B), B, 0, stream>>>(ssrc, sdst, emaxU, emaxF, denom, N);
        k_att_ee<<<cdiv(E, B), B, 0, stream>>>(src, dst, ssrc, sdst, emaxF, ee, denom, E);
        k_att_scatter<<<cdiv((long long)E * 32, B), B, 0, stream>>>(src, dst, ee, denom, HK,
                                                                    OUT, E);
        k_att_self<<<cdiv((long long)N * 32, B), B, 0, stream>>>(ssrc, sdst, emaxF, denom, HK,
                                                                 OUT, N);
    }

    // ---- x3 = OUT/HEADS + b3 (into d_out) ----
    k_x3<<<cdiv((long long)N * HID, B), B, 0, stream>>>(OUT, b3, x3, N);

    // ---- global attention pool ----
    float* T = OUT;  // reuse
    {
        int waves = (N / 16) * (HID / 16);
        k_gemm_wmma_f32<<<cdiv(waves * 32, B), B, 0, stream>>>(x3, HID, Gw1, HID, 0, T, HID,
                                                               Gb1, 1, N, HID, HID);
    }
    k_rowdot<<<cdiv((long long)N * 32, B), B, 0, stream>>>(T, Gw2, Gb2, gate, N);
    k_scal_init<<<1, 32, 0, stream>>>(scal);
    k_gmax<<<cdiv(N, B), B, 0, stream>>>(gate, scal, N);
    k_wts<<<cdiv(N, B), B, 0, stream>>>(gate, scal, wts, N);
    k_fill<<<1, 256, 0, stream>>>(gacc, 0.0f, HID);
    k_gacc<<<cdiv(N, 128), 256, 0, stream>>>(wts, scal, x3, gacc, N);
    k_gfinal<<<1, 256, 0, stream>>>(gacc, Mw, Mb, gO);
}